// DTCWT_28097676051075
// MI455X (gfx1250) — compile-verified
//
#include <hip/hip_runtime.h>

// ---------------------------------------------------------------------------
// DTCWT (dual-tree complex wavelet transform), level 3, on gfx1250 (MI455X).
//
// Circular 10-tap analysis filter + downsample-by-2 along one axis:
//     out[m] = sum_t f[t] * x[(2m + 5 - t) mod N]
// Block of 16 outputs == banded 16x40 matrix A (A[o,r]=f[2o+9-r]) times a
// 40x16 input patch -> 10 chunks of V_WMMA_F32_16X16X4_F32 (f32 WMMA,
// wave32). Each wave computes one dual (lo,hi) 16x16 tile sharing B.
// The per-lane A operands are pre-expanded into small global tables so the
// hot loop is: load A (b128, coalesced) + load 2x B rows + 2x WMMA.
// ---------------------------------------------------------------------------

typedef __attribute__((ext_vector_type(2))) float v2f;
typedef __attribute__((ext_vector_type(8))) float v8f;

// ---------------------------------------------------------------------------
// A-operand table builder.  Table t (0..5), 10 chunks x 32 lanes x float4:
//   {A_lo[K=kb], A_lo[K=kb+1], A_hi[K=kb], A_hi[K=kb+1]} for lane's row M.
//  t=0/1: Faf[m]*0.5 (level-0 rows)   t=2/3: Faf[n] (level-0 cols)
//  t=4/5: af[m or n]                  (levels 1-2, rows & cols)
// ---------------------------------------------------------------------------
__global__ __launch_bounds__(320) void build_atab_kernel(
        const float* __restrict__ Faf, const float* __restrict__ af,
        float4* __restrict__ tab) {
    const int t  = blockIdx.x;          // table id 0..5
    const int id = threadIdx.x;         // c*32 + lane
    const int c = id >> 5, lane = id & 31;

    const float* f = (t < 4) ? (Faf + (t & 1) * 20) : (af + (t & 1) * 20);
    const float scale = (t < 2) ? 0.5f : 1.0f;   // fold x/2 into level-0 rows

    const int M  = lane & 15;
    const int kb = (lane < 16) ? 0 : 2;
    const int r0 = 4 * c + kb;
    const int ta = 2 * M + 9 - r0;      // A[o,r] = f[2o+9-r] if 0<=..<=9
    const int tb = ta - 1;
    const int tac = (ta < 0) ? 0 : (ta > 9 ? 9 : ta);
    const int tbc = (tb < 0) ? 0 : (tb > 9 ? 9 : tb);
    const float ma = (ta == tac) ? scale : 0.0f;
    const float mb = (tb == tbc) ? scale : 0.0f;

    float4 v;
    v.x = f[tac]      * ma;   // lo filter
    v.y = f[tbc]      * mb;
    v.z = f[10 + tac] * ma;   // hi filter
    v.w = f[10 + tbc] * mb;
    tab[t * 320 + id] = v;
}

// ---------------------------------------------------------------------------
struct AfbParams {
    const float* src;
    float* dlo;
    float* dhi;
    const float4* atab;               // 320 float4 (10 chunks x 32 lanes)
    int N;                            // filtered-axis length (input)
    int tiles_m, tiles_n;             // (N/2)/16, free-count/16
    int outer_c, outer_b;             // 1 or 3 (chan), batch=8
    int s_r, s_n, s_ob, s_oc;         // src strides (elements, 32-bit safe)
    int dl_r, dl_n, dl_ob, dl_oc;     // dst-lo strides
    int dh_r, dh_n, dh_ob, dh_oc;     // dst-hi strides
};

#define WAVES_PER_BLOCK 4

__device__ __forceinline__ v8f wmma_f32(v2f a, v2f b, v8f c) {
    return __builtin_amdgcn_wmma_f32_16x16x4_f32(
        false, a, false, b, (short)0, c, false, false);
}

__global__ __launch_bounds__(128) void afb_wmma_kernel(AfbParams p) {
    const int tid  = threadIdx.x;
    const int wave = tid >> 5;
    const int lane = tid & 31;
    const long tile = (long)blockIdx.x * WAVES_PER_BLOCK + wave;
    const long tiles_total =
        (long)p.outer_b * p.outer_c * p.tiles_m * p.tiles_n;
    if (tile >= tiles_total) return;     // wave-uniform: EXEC stays all-ones

    const int  nt = (int)(tile % p.tiles_n);
    long t1 = tile / p.tiles_n;
    const int  mt = (int)(t1 % p.tiles_m);
    long t2 = t1 / p.tiles_m;
    const int  oc = (int)(t2 % p.outer_c);
    const int  ob = (int)(t2 / p.outer_c);

    // ISA operand layouts (wave32):
    //  A 16x4 f32 : lane L holds row M=L%16; VGPR0/1 = K=kb, kb+1, kb=(L<16?0:2)
    //  B 4x16 f32 : lane L holds col N=L%16; VGPR0/1 = K=kb, kb+1
    //  C/D 16x16  : VGPR v, lanes 0-15 -> M=v, lanes 16-31 -> M=v+8; N=L%16
    const int kb = (lane < 16) ? 0 : 2;
    const int nl = lane & 15;
    const int m0 = mt * 16;
    const int n0 = nt * 16;

    const int src_off = ob * p.s_ob + oc * p.s_oc + (n0 + nl) * p.s_n;
    const int rbase   = 2 * m0 - 4;      // input row for band index r=0

    // Pre-expanded A operands: 10 coalesced b128 loads (table is L2-hot)
    v2f a_lo[10], a_hi[10];
#pragma unroll
    for (int c = 0; c < 10; ++c) {
        const float4 a = p.atab[c * 32 + lane];
        a_lo[c].x = a.x; a_lo[c].y = a.y;
        a_hi[c].x = a.z; a_hi[c].y = a.w;
    }

    v8f acc_lo = {0.f,0.f,0.f,0.f,0.f,0.f,0.f,0.f};
    v8f acc_hi = {0.f,0.f,0.f,0.f,0.f,0.f,0.f,0.f};

    if (rbase >= 0 && rbase + 40 <= p.N) {
        // Fast path (all interior m-tiles): no circular wrap, incremental
        // 32-bit offsets, prefetch two K-chunks ahead (global_prefetch_b8).
        int ofs = src_off + (rbase + kb) * p.s_r;
        const int step = 4 * p.s_r;
#pragma unroll
        for (int c = 0; c < 10; ++c) {
            v2f b;
            b.x = p.src[ofs];
            b.y = p.src[ofs + p.s_r];
            if (c < 8)
                __builtin_prefetch(p.src + (ofs + 2 * step), 0, 3);
            ofs += step;
            acc_lo = wmma_f32(a_lo[c], b, acc_lo);
            acc_hi = wmma_f32(a_hi[c], b, acc_hi);
        }
    } else {
        // Edge m-tiles: per-row circular wrap via VALU selects (no EXEC use)
#pragma unroll
        for (int c = 0; c < 10; ++c) {
            int rg0 = rbase + 4 * c + kb;
            int rg1 = rg0 + 1;
            rg0 = (rg0 < 0) ? rg0 + p.N : ((rg0 >= p.N) ? rg0 - p.N : rg0);
            rg1 = (rg1 < 0) ? rg1 + p.N : ((rg1 >= p.N) ? rg1 - p.N : rg1);
            v2f b;
            b.x = p.src[src_off + rg0 * p.s_r];
            b.y = p.src[src_off + rg1 * p.s_r];
            acc_lo = wmma_f32(a_lo[c], b, acc_lo);
            acc_hi = wmma_f32(a_hi[c], b, acc_hi);
        }
    }

    // Store D per the 16x16 f32 C/D layout
    const int mb = m0 + ((lane < 16) ? 0 : 8);
    int lo_off = ob * p.dl_ob + oc * p.dl_oc + (n0 + nl) * p.dl_n + mb * p.dl_r;
    int hi_off = ob * p.dh_ob + oc * p.dh_oc + (n0 + nl) * p.dh_n + mb * p.dh_r;
#pragma unroll
    for (int v = 0; v < 8; ++v) {
        p.dlo[lo_off] = acc_lo[v];
        p.dhi[hi_off] = acc_hi[v];
        lo_off += p.dl_r;
        hi_off += p.dh_r;
    }
}

// ---------------------------------------------------------------------------
// In-place dual-tree combine on d_out: pair (0,0)&(1,1) quadrants (pair 0)
// and (0,1)&(1,0) (pair 1): p,q <- (p+q)/sqrt2, (p-q)/sqrt2.  Skips each
// quadrant's 64x64 lowpass corner.  float4 over the contiguous w*c dim.
// ---------------------------------------------------------------------------
__global__ __launch_bounds__(256) void combine_kernel(float* out) {
    const float k = 0.70710678118654752440f;
    const int pair = blockIdx.y;
    const long idx = (long)blockIdx.x * blockDim.x + threadIdx.x;
    const long total = 8L * 512 * 384;          // batches * rows * float4/row
    if (idx >= total) return;
    const int col4 = (int)(idx % 384);          // quadrant-local, float4 units
    const int row  = (int)((idx / 384) % 512);  // quadrant-local row
    const int b    = (int)(idx / (384L * 512));
    if (row < 64 && col4 < 48) return;          // 64x64x3 lowpass corner

    float4* o4 = (float4*)out;
    const long base = (long)b * 786432 + (long)row * 768 + col4;
    const long po = (pair == 0) ? 0L      : 384L;     // (0,0) / (0,1)
    const long qo = (pair == 0) ? 393600L : 393216L;  // (1,1) / (1,0)

    const float4 pv = o4[base + po];
    const float4 qv = o4[base + qo];
    float4 rs, rd;
    rs.x = (pv.x + qv.x) * k;  rd.x = (pv.x - qv.x) * k;
    rs.y = (pv.y + qv.y) * k;  rd.y = (pv.y - qv.y) * k;
    rs.z = (pv.z + qv.z) * k;  rd.z = (pv.z - qv.z) * k;
    rs.w = (pv.w + qv.w) * k;  rd.w = (pv.w - qv.w) * k;
    o4[base + po] = rs;
    o4[base + qo] = rd;
}

// ---------------------------------------------------------------------------
extern "C" void kernel_launch(void* const* d_in, const int* in_sizes, int n_in,
                              void* d_out, int out_size, void* d_ws, size_t ws_size,
                              hipStream_t stream) {
    (void)in_sizes; (void)n_in; (void)out_size; (void)ws_size;
    const float* x   = (const float*)d_in[0];   // (8,512,512,3)
    const float* Faf = (const float*)d_in[1];   // (2,2,10)
    const float* af  = (const float*)d_in[2];   // (2,2,10)
    float* out = (float*)d_out;                 // (8,1024,1024,3)
    float* ws  = (float*)d_ws;

    // Workspace layout (floats)
    float* Lr   = ws;                 // (8,256,512,3) = 3,145,728
    float* Hr   = Lr + 3145728;       // (8,256,512,3)
    float* lo0  = Hr + 3145728;       // (8,256,256,3) = 1,572,864
    float* lo1  = lo0 + 1572864;      // (8,128,128,3) =   393,216
    float* atab = lo1 + 393216;       // 6 tables x 1280 floats = 7,680

    const int OUT_B = 3145728;        // batch stride in d_out (floats)
    const int OUT_H = 3072;           // row stride in d_out

    // Expand filters into per-lane WMMA A-operand tables (tiny, L2-hot)
    build_atab_kernel<<<6, 320, 0, stream>>>(Faf, af, (float4*)atab);

    auto tab = [&](int t) { return (const float4*)(atab + t * 1280); };
    auto launch_afb = [&](const AfbParams& p) {
        const long tiles = (long)p.outer_b * p.outer_c * p.tiles_m * p.tiles_n;
        const int blocks = (int)((tiles + WAVES_PER_BLOCK - 1) / WAVES_PER_BLOCK);
        afb_wmma_kernel<<<blocks, 128, 0, stream>>>(p);
    };

    for (int m = 0; m < 2; ++m)
    for (int n = 0; n < 2; ++n) {
        const long qoff = (long)m * 512 * OUT_H + (long)n * 512 * 3;
        const float* src = x;
        for (int j = 0; j < 3; ++j) {
            const int Hs = 512 >> j, Ws = 512 >> j;
            const int Hs2 = Hs / 2, Ws2 = Ws / 2;

            // ---- Row pass: src (8,Hs,Ws,3) -> Lr,Hr (8,Hs2,Ws,3) ----
            AfbParams rp = {};
            rp.src = src; rp.dlo = Lr; rp.dhi = Hr;
            rp.atab = (j == 0) ? tab(m) : tab(4 + m);
            rp.N = Hs; rp.tiles_m = Hs2 / 16; rp.tiles_n = (Ws * 3) / 16;
            rp.outer_c = 1; rp.outer_b = 8;
            rp.s_r = Ws * 3; rp.s_n = 1; rp.s_ob = Hs * Ws * 3; rp.s_oc = 0;
            rp.dl_r = Ws * 3; rp.dl_n = 1; rp.dl_ob = Hs2 * Ws * 3; rp.dl_oc = 0;
            rp.dh_r = rp.dl_r; rp.dh_n = rp.dl_n;
            rp.dh_ob = rp.dl_ob; rp.dh_oc = rp.dl_oc;
            launch_afb(rp);

            // ---- Col passes: (8,Hs2,Ws,3) -> (8,Hs2,Ws2,3), filter along W ----
            AfbParams cp = {};
            cp.atab = (j == 0) ? tab(2 + n) : tab(4 + n);
            cp.N = Ws; cp.tiles_m = Ws2 / 16; cp.tiles_n = Hs2 / 16;
            cp.outer_c = 3; cp.outer_b = 8;
            cp.s_r = 3; cp.s_n = Ws * 3; cp.s_ob = Hs2 * Ws * 3; cp.s_oc = 1;

            // Col pass on Lr: lo -> next-level buffer (or final lo slot),
            //                 hi -> subband s0 slot at (0, Hs2)
            {
                AfbParams p = cp;
                p.src = Lr;
                if (j < 2) {
                    p.dlo = (j == 0) ? lo0 : lo1;
                    p.dl_r = 3; p.dl_n = Ws2 * 3;
                    p.dl_ob = Hs2 * Ws2 * 3; p.dl_oc = 1;
                } else {
                    p.dlo = out + qoff;                    // lo slot (0,0) 64x64
                    p.dl_r = 3; p.dl_n = OUT_H; p.dl_ob = OUT_B; p.dl_oc = 1;
                }
                p.dhi = out + qoff + Hs2 * 3;              // s0 slot (0,Hs2)
                p.dh_r = 3; p.dh_n = OUT_H; p.dh_ob = OUT_B; p.dh_oc = 1;
                launch_afb(p);
            }
            // Col pass on Hr: lo -> s1 slot (Hs2,0), hi -> s2 slot (Hs2,Hs2)
            {
                AfbParams p = cp;
                p.src = Hr;
                p.dlo = out + qoff + (long)Hs2 * OUT_H;
                p.dl_r = 3; p.dl_n = OUT_H; p.dl_ob = OUT_B; p.dl_oc = 1;
                p.dhi = out + qoff + (long)Hs2 * OUT_H + Hs2 * 3;
                p.dh_r = 3; p.dh_n = OUT_H; p.dh_ob = OUT_B; p.dh_oc = 1;
                launch_afb(p);
            }

            src = (j == 0) ? lo0 : lo1;
        }
    }

    // Dual-tree combine, in place on d_out (2 quadrant pairs)
    {
        const long total = 8L * 512 * 384;
        dim3 grid((unsigned)((total + 255) / 256), 2);
        combine_kernel<<<grid, 256, 0, stream>>>(out);
    }
}